// MHCLiteResidualMixing_257698038391
// MI455X (gfx1250) — compile-verified
//
#include <hip/hip_runtime.h>
#include <hip/hip_bf16.h>

typedef float v2f __attribute__((ext_vector_type(2)));
typedef float v8f __attribute__((ext_vector_type(8)));

#define NS 4          // n_streams
#define DDIM 1024     // D
#define SINK_ITERS 20

// ---------------------------------------------------------------------------
// Stage 1: Sinkhorn normalization of the 4x4 H_raw -> doubly stochastic H.
// ~1K flops total: one lane does it serially, result lands in d_ws.
// ---------------------------------------------------------------------------
__global__ void mhc_sinkhorn_kernel(const float* __restrict__ Hraw,
                                    float* __restrict__ Hout) {
  if (threadIdx.x == 0) {
    float A[NS][NS];
    #pragma unroll
    for (int i = 0; i < NS; ++i)
      #pragma unroll
      for (int j = 0; j < NS; ++j)
        A[i][j] = fabsf(Hraw[i * NS + j]) + 1e-8f;

    for (int it = 0; it < SINK_ITERS; ++it) {
      #pragma unroll
      for (int i = 0; i < NS; ++i) {
        float s = A[i][0] + A[i][1] + A[i][2] + A[i][3];
        #pragma unroll
        for (int j = 0; j < NS; ++j) A[i][j] = A[i][j] / s;
      }
      #pragma unroll
      for (int j = 0; j < NS; ++j) {
        float s = A[0][j] + A[1][j] + A[2][j] + A[3][j];
        #pragma unroll
        for (int i = 0; i < NS; ++i) A[i][j] = A[i][j] / s;
      }
    }
    #pragma unroll
    for (int i = 0; i < NS * NS; ++i) Hout[i] = A[i >> 2][i & 3];
  }
}

// ---------------------------------------------------------------------------
// Stage 2: out[r,n,d] = sum_m H[n,m] * x[r,m,d]   (r = b*T+t, 16384 rows)
// Bandwidth-bound (512 MB traffic @ 23.3 TB/s ~ 22us); compute via
// V_WMMA_F32_16X16X4_F32 with K = n_streams = 4:
//   A (16x4)  = H padded (rows 0-3), documented layout:
//               lanes 0-15: v0=A[m][0], v1=A[m][1]; lanes 16-31: K=2,3
//   B (4x16)  = x slab, 16 consecutive d-columns x 4 streams (K0|K2 / K1|K3
//               split mirroring A's documented layout)
//   D rows 0-3 = 4 output streams at those 16 d-columns (documented C/D map)
// Two adjacent 16-wide tiles per iteration; NT hints (streams exceed L2).
// ---------------------------------------------------------------------------
__global__ void mhc_mix_wmma_kernel(const float* __restrict__ x,
                                    const float* __restrict__ H,
                                    float* __restrict__ out,
                                    int nPairs) {
  const int lane   = threadIdx.x & 31;
  const int wave   = (int)((blockIdx.x * blockDim.x + threadIdx.x) >> 5);
  const int nwaves = (int)((gridDim.x * blockDim.x) >> 5);

  // Build A = H zero-padded to 16x4 (per-wave constant).
  const int m  = lane & 15;
  const int kb = (lane >> 4) << 1;   // 0 for lanes 0-15, 2 for lanes 16-31
  v2f a;
  a.x = (m < NS) ? H[m * NS + kb]     : 0.0f;
  a.y = (m < NS) ? H[m * NS + kb + 1] : 0.0f;

  const int col = lane & 15;         // d-column within tile
  const int s0  = (lane >> 4) << 1;  // stream feeding b.x: 0 (low half) / 2 (high half)

  for (int p = wave; p < nPairs; p += nwaves) {
    const int r    = p >> 5;               // row index in [0, B*T)
    const int d0   = (p & 31) << 5;        // 32-column chunk within D
    const int base = (r << 12) + d0 + col; // r*NS*DDIM + d0 + col

    v2f b0, b1;
    b0.x = __builtin_nontemporal_load(&x[base + s0 * DDIM]);
    b0.y = __builtin_nontemporal_load(&x[base + (s0 + 1) * DDIM]);
    b1.x = __builtin_nontemporal_load(&x[base + 16 + s0 * DDIM]);
    b1.y = __builtin_nontemporal_load(&x[base + 16 + (s0 + 1) * DDIM]);

    v8f c = {};
    v8f dA = __builtin_amdgcn_wmma_f32_16x16x4_f32(false, a, false, b0,
                                                   (short)0, c, false, false);
    v8f dB = __builtin_amdgcn_wmma_f32_16x16x4_f32(false, a, false, b1,
                                                   (short)0, c, false, false);

    // D rows 0-3 live in VGPRs 0-3 of lanes 0-15 (N = lane). 64B coalesced
    // stores per stream row.
    if (lane < 16) {
      #pragma unroll
      for (int n = 0; n < NS; ++n) {
        __builtin_nontemporal_store(dA[n], &out[base + n * DDIM]);
        __builtin_nontemporal_store(dB[n], &out[base + 16 + n * DDIM]);
      }
    }
  }
}

extern "C" void kernel_launch(void* const* d_in, const int* in_sizes, int n_in,
                              void* d_out, int out_size, void* d_ws, size_t ws_size,
                              hipStream_t stream) {
  const float* x    = (const float*)d_in[0];   // [B,T,NS,D] f32
  const float* Hraw = (const float*)d_in[1];   // [NS,NS]    f32
  float* out = (float*)d_out;                  // [B,T,NS,D] f32
  float* H   = (float*)d_ws;                   // 16 floats scratch

  // Stage 1: tiny Sinkhorn, one wave.
  mhc_sinkhorn_kernel<<<1, 32, 0, stream>>>(Hraw, H);

  // Stage 2: streaming WMMA mix.
  const int R      = in_sizes[0] / (NS * DDIM);     // B*T rows
  const int nPairs = R * (DDIM / 32);               // 32-column chunks
  const int threads = 256;                          // 8 waves/block
  const int blocks  = 2048;                         // 16384 waves total
  mhc_mix_wmma_kernel<<<blocks, threads, 0, stream>>>(x, H, out, nPairs);
}